// RejectionSampler_87205015978231
// MI455X (gfx1250) — compile-verified
//
#include <hip/hip_runtime.h>
#include <cstdint>

// ---------------------------------------------------------------------------
// Rejection sampler (greedy spec-decode accept):
//   Phase 1: argmax over vocab for each of B*(K+1)=1152 rows (V=128000 fp32)
//            -> pure streaming reduction, 590 MB once, ~25us @ 23.3 TB/s
//   Phase 2: prefix-accept vs draft ids, emit tokens / -1 padding (tiny)
//
// Phase 1 uses the CDNA5 async global->LDS streaming path
// (GLOBAL_LOAD_ASYNC_TO_LDS_B128 + ASYNCcnt, double-buffered 2x32KB in LDS).
// ---------------------------------------------------------------------------

#define BLOCK 512                       // 16 waves (wave32)
#define TILE  8192                      // floats per LDS tile (32 KB)
#define CHUNKS_PER_THREAD ((TILE / 4) / BLOCK)   // 4 x float4 per thread/tile
#define NWAVES (BLOCK / 32)

// One per-lane 16-byte async copy: from global address base plus gOff to LDS offset ldsOff, NT hint
// (read-once stream of 590MB > 192MB L2: do not cache).
__device__ __forceinline__ void async_copy_b128(uint32_t ldsOff, uint32_t gOff,
                                                const void* base) {
  asm volatile("global_load_async_to_lds_b128 %0, %1, %2 th:TH_LOAD_NT"
               :: "v"(ldsOff), "v"(gOff), "s"(base)
               : "memory");
}
__device__ __forceinline__ void wait_async_le4() {
  asm volatile("s_wait_asynccnt 0x4" ::: "memory");
}
__device__ __forceinline__ void wait_async_0() {
  asm volatile("s_wait_asynccnt 0x0" ::: "memory");
}

// Issue one full tile (no clamping needed; tile entirely in-bounds).
__device__ __forceinline__ void issue_tile_full(uint32_t ldsBase, uint32_t tb,
                                                int t, const void* rowBase) {
#pragma unroll
  for (int k = 0; k < CHUNKS_PER_THREAD; ++k) {
    const uint32_t c = (uint32_t)t + (uint32_t)k * BLOCK;
    async_copy_b128(ldsBase + c * 16u, tb + c * 16u, rowBase);
  }
}

// Issue the (possibly partial) tail tile: every wave still issues exactly
// CHUNKS_PER_THREAD ops (clamped addresses; junk filtered during reduce)
// so per-wave ASYNCcnt bookkeeping stays uniform.
__device__ __forceinline__ void issue_tile_tail(uint32_t ldsBase, uint32_t tb,
                                                int t, const void* rowBase,
                                                uint32_t maxOff) {
#pragma unroll
  for (int k = 0; k < CHUNKS_PER_THREAD; ++k) {
    const uint32_t c = (uint32_t)t + (uint32_t)k * BLOCK;
    uint32_t g = tb + c * 16u;
    if (g > maxOff) g = maxOff;
    async_copy_b128(ldsBase + c * 16u, g, rowBase);
  }
}

__global__ __launch_bounds__(BLOCK)
void argmax_rows_kernel(const float* __restrict__ logits, int V,
                        int* __restrict__ out_tok) {
  __shared__ __align__(16) float smem[2][TILE];   // 64 KB double buffer
  __shared__ float s_wval[NWAVES];
  __shared__ int   s_widx[NWAVES];

  const int row = blockIdx.x;
  const int t   = threadIdx.x;

  const float*   rowBase  = logits + (size_t)row * (size_t)V;
  const uint32_t maxOff   = (uint32_t)V * 4u - 16u;
  const int      numTiles = (V + TILE - 1) / TILE;

  // Flat address of __shared__ truncated to 32 bits == wave-relative LDS
  // byte offset (flat LDS aperture keeps the offset in addr[31:0]).
  const uint32_t ldsBase0 = (uint32_t)(uintptr_t)&smem[0][0];
  const uint32_t ldsBase1 = (uint32_t)(uintptr_t)&smem[1][0];

  // Prologue: stage tile 0 into buffer 0.
  if (numTiles > 1) issue_tile_full(ldsBase0, 0u, t, rowBase);
  else              issue_tile_tail(ldsBase0, 0u, t, rowBase, maxOff);

  float best    = -3.402823466e38f;
  int   bestIdx = 0;

  for (int i = 0; i < numTiles; ++i) {
    // Kick off tile i+1 into the other buffer, then wait for tile i's
    // 4 per-wave copies (ASYNCcnt retires in order -> wait <= 4).
    if (i + 1 < numTiles) {
      const uint32_t tb = (uint32_t)(i + 1) * (uint32_t)TILE * 4u;
      const uint32_t lb = ((i + 1) & 1) ? ldsBase1 : ldsBase0;
      if (i + 2 < numTiles) issue_tile_full(lb, tb, t, rowBase);
      else                  issue_tile_tail(lb, tb, t, rowBase, maxOff);
      wait_async_le4();
    } else {
      wait_async_0();
    }
    __syncthreads();                       // tile i resident for all waves

    const float* buf       = (i & 1) ? smem[1] : smem[0];
    const int    tileStart = i * TILE;
    const bool   fullTile  = (tileStart + TILE) <= V;

    if (fullTile) {
#pragma unroll
      for (int k = 0; k < CHUNKS_PER_THREAD; ++k) {
        const int    c  = t + k * BLOCK;
        const float4 v4 = *reinterpret_cast<const float4*>(buf + c * 4);
        const int    gi = tileStart + c * 4;
        const float  vv[4] = {v4.x, v4.y, v4.z, v4.w};
#pragma unroll
        for (int e = 0; e < 4; ++e) {
          const float v = vv[e];
          // first-occurrence argmax: strictly greater, or equal & lower idx
          if (v > best || (v == best && (gi + e) < bestIdx)) {
            best = v;
            bestIdx = gi + e;
          }
        }
      }
    } else {
#pragma unroll
      for (int k = 0; k < CHUNKS_PER_THREAD; ++k) {
        const int    c  = t + k * BLOCK;
        const float4 v4 = *reinterpret_cast<const float4*>(buf + c * 4);
        const int    gi = tileStart + c * 4;
        const float  vv[4] = {v4.x, v4.y, v4.z, v4.w};
#pragma unroll
        for (int e = 0; e < 4; ++e) {
          const int   gidx = gi + e;
          const float v    = vv[e];
          if (gidx < V && (v > best || (v == best && gidx < bestIdx))) {
            best = v;
            bestIdx = gidx;
          }
        }
      }
    }
    // Protect buf[i&1] only if a future issue (tile i+2) will overwrite it.
    if (i + 2 < numTiles) __syncthreads();
  }

  // Wave32 butterfly reduction (val, idx) with min-index tie-break.
#pragma unroll
  for (int off = 16; off > 0; off >>= 1) {
    const float ov = __shfl_xor(best, off, 32);
    const int   oi = __shfl_xor(bestIdx, off, 32);
    if (ov > best || (ov == best && oi < bestIdx)) { best = ov; bestIdx = oi; }
  }
  const int wave = t >> 5;
  if ((t & 31) == 0) { s_wval[wave] = best; s_widx[wave] = bestIdx; }
  __syncthreads();
  if (t == 0) {
    float b  = s_wval[0];
    int   bi = s_widx[0];
#pragma unroll
    for (int w = 1; w < NWAVES; ++w) {
      const float ov = s_wval[w];
      const int   oi = s_widx[w];
      if (ov > b || (ov == b && oi < bi)) { b = ov; bi = oi; }
    }
    out_tok[row] = bi;
  }
}

// Phase 2: accepted-prefix length + emit tokens (tiny, 128 threads).
__global__ void finalize_kernel(const int* __restrict__ tok,
                                const int* __restrict__ draft,
                                float* __restrict__ out,
                                int batch, int K) {
  const int b = blockIdx.x * blockDim.x + threadIdx.x;
  if (b >= batch) return;
  int n = 0;
  for (int j = 0; j < K; ++j) {
    if (draft[b * K + j] == tok[b * (K + 1) + j]) ++n;
    else break;
  }
  for (int j = 0; j <= K; ++j) {
    out[b * (K + 1) + j] = (j <= n) ? (float)tok[b * (K + 1) + j] : -1.0f;
  }
}

extern "C" void kernel_launch(void* const* d_in, const int* in_sizes, int n_in,
                              void* d_out, int out_size, void* d_ws, size_t ws_size,
                              hipStream_t stream) {
  const float* logits = (const float*)d_in[0];
  const int*   draft  = (const int*)d_in[1];    // integer input -> int32 per harness

  const int K     = 8;
  const int batch = in_sizes[1] / K;            // 128
  const int rows  = batch * (K + 1);            // 1152
  const int V     = in_sizes[0] / rows;         // 128000

  int* tok = (int*)d_ws;                        // rows * 4 bytes of scratch

  argmax_rows_kernel<<<rows, BLOCK, 0, stream>>>(logits, V, tok);
  finalize_kernel<<<(batch + 127) / 128, 128, 0, stream>>>(
      tok, draft, (float*)d_out, batch, K);
}